// MultiScaleRetention_4131758539027
// MI455X (gfx1250) — compile-verified
//
#include <hip/hip_runtime.h>
#include <math.h>
#include <stdint.h>

typedef __attribute__((ext_vector_type(2))) float v2f;
typedef __attribute__((ext_vector_type(8))) float v8f;
typedef __attribute__((ext_vector_type(4))) unsigned int v4u;
typedef __attribute__((ext_vector_type(8))) unsigned int v8u;

#define BB    8
#define EE    256
#define HH    8
#define KDIM  32
#define LL    1024
#define SCALING 0.17677669529663687f  /* 32^-0.5 */
#define LN_EPS  1e-5f
#define SROW  1028                    /* padded LDS row stride (floats) */

__device__ __forceinline__ float wave_reduce_add(float x) {
    #pragma unroll
    for (int m = 16; m >= 1; m >>= 1) x += __shfl_xor(x, m, 32);
    return x;
}

__device__ __forceinline__ int iabs(int a) { return a < 0 ? -a : a; }

__device__ __forceinline__ float decay_of(int h) {
    return logf(1.0f - exp2f(-2.0f - 5.0f * (float)h / 8.0f));
}

// ---- Tensor Data Mover: DMA a 2D fp32 tile (tile1 rows x tile0 cols) into LDS ----
// D# group0/group1 built per CDNA5 ISA 8.3/8.4; 2-group form (VADDR2/3 = NULL).
__device__ __forceinline__ void tdm_load_2d_f32(uint32_t lds_off, const float* gptr,
                                                uint32_t tile0, uint32_t tile1,
                                                uint32_t stride0) {
    uint64_t ga = (uint64_t)(uintptr_t)gptr;
    v4u g0;
    g0[0] = 1u;                                   // count=1, user descriptor
    g0[1] = lds_off;                              // lds_addr (bytes)
    g0[2] = (uint32_t)ga;                         // global_addr[31:0]
    g0[3] = (uint32_t)((ga >> 32) & 0x1FFFFFFu)   // global_addr[56:32]
            | (2u << 30);                         // type=2 (image)
    v8u g1;
    g1[0] = 0x20000u;                             // data_size=2 (4 bytes), no mask
    g1[1] = (tile0 & 0xFFFFu) << 16;              // tensor_dim0[15:0]  (== tile0)
    g1[2] = (tile0 >> 16) | ((tile1 & 0xFFFFu) << 16); // tensor_dim0 hi | tensor_dim1 lo
    g1[3] = (tile1 >> 16) | ((tile0 & 0xFFFFu) << 16); // tensor_dim1 hi | tile_dim0
    g1[4] = tile1 & 0xFFFFu;                      // tile_dim1, tile_dim2=0
    g1[5] = stride0;                              // tensor_dim0_stride[31:0]
    g1[6] = 0u;                                   // stride0 hi | stride1 lo
    g1[7] = 0u;                                   // stride1 hi
    asm volatile("tensor_load_to_lds %0, %1" :: "s"(g0), "s"(g1) : "memory");
}

// ---------------- rotary tables: sin/cos (L x 32) ----------------
__global__ void k_tables(float* __restrict__ sinT, float* __restrict__ cosT) {
    int tid = blockIdx.x * blockDim.x + threadIdx.x;   // 32768
    int l = tid >> 5, dd = tid & 31;
    int i = dd >> 1;                                   // repeated pairs
    float ang = powf(10000.0f, -(float)i / 15.0f);
    float idx = (float)((l >> 5) + (l & 31));          // i1 + i2
    sinT[tid] = sinf(idx * ang);
    cosT[tid] = cosf(idx * ang);
}

// -------- mask row-sum normalizer: rmask[h][q] = rsqrt(sum_k exp(decay*d)) --------
__global__ void k_rowsum(float* __restrict__ rmask) {
    int w = threadIdx.x >> 5, lane = threadIdx.x & 31;
    int gid = blockIdx.x * 8 + w;                      // 0..8191
    int h = gid >> 10, q = gid & 1023;
    float dec = decay_of(h);
    int q1 = q >> 5, q2 = q & 31;
    float s = 0.0f;
    for (int j = 0; j < 32; ++j) {
        int k = lane + 32 * j;
        float d = (float)(iabs(q1 - (k >> 5)) + iabs(q2 - (k & 31)));
        s += __expf(dec * d);
    }
    s = wave_reduce_add(s);
    if (lane == 0) rmask[gid] = rsqrtf(s);
}

// -------- fused QKV projection + scaling + theta-shift rotary --------
// A tile (16 x 256 of x) is shared by all 8 waves of the block: TDM it into LDS once.
__global__ void __launch_bounds__(256) k_proj(
        const float* __restrict__ x,  const float* __restrict__ Wq,
        const float* __restrict__ Wk, const float* __restrict__ Wv,
        const float* __restrict__ sinT, const float* __restrict__ cosT,
        float* __restrict__ qr, float* __restrict__ kr, float* __restrict__ vvT) {
    __shared__ float xs[16 * EE];                       // 16KB A tile
    int w = threadIdx.x >> 5, lane = threadIdx.x & 31;
    int lh = lane >> 4, lr = lane & 15;
    int mt = blockIdx.x >> 1;                           // shared across block
    int nt = (blockIdx.x & 1) * 8 + w;

    if (threadIdx.x < 32) {                             // wave 0 issues the DMA
        tdm_load_2d_f32((uint32_t)(uintptr_t)&xs[0], x + (size_t)mt * 16 * EE,
                        EE, 16, EE);
        __builtin_amdgcn_s_wait_tensorcnt(0);
    }
    __syncthreads();

    const float* wqrow = Wq + (size_t)(nt * 16 + lr) * EE;
    const float* wkrow = Wk + (size_t)(nt * 16 + lr) * EE;
    const float* wvrow = Wv + (size_t)(nt * 16 + lr) * EE;
    v8f cq = {}; v8f ck = {}; v8f cv = {};
    for (int k0 = 0; k0 < EE; k0 += 4) {
        int ko = k0 + 2 * lh;
        v2f a  = *reinterpret_cast<const v2f*>(&xs[lr * EE + ko]);
        v2f bq = *reinterpret_cast<const v2f*>(wqrow + ko);
        v2f bk = *reinterpret_cast<const v2f*>(wkrow + ko);
        v2f bv = *reinterpret_cast<const v2f*>(wvrow + ko);
        cq = __builtin_amdgcn_wmma_f32_16x16x4_f32(false, a, false, bq, (short)0, cq, false, false);
        ck = __builtin_amdgcn_wmma_f32_16x16x4_f32(false, a, false, bk, (short)0, ck, false, false);
        cv = __builtin_amdgcn_wmma_f32_16x16x4_f32(false, a, false, bv, (short)0, cv, false, false);
    }
    int col = nt * 16 + lr;
    int h = col >> 5, d = col & 31;
    float sgn = (d & 1) ? 1.0f : -1.0f;                 // rotate_every_two sign
    #pragma unroll
    for (int r = 0; r < 8; ++r) {
        int row = mt * 16 + r + 8 * lh;                 // global (b*L + l)
        int b = row >> 10, l = row & 1023;
        float sn = sinT[l * 32 + d], cs = cosT[l * 32 + d];
        float qv = cq[r];
        float kv = ck[r] * SCALING;
        float qp = __shfl_xor(qv, 1, 32);               // pair partner (d^1)
        float kp = __shfl_xor(kv, 1, 32);
        int bh = b * HH + h;
        size_t o = ((size_t)bh * LL + l) * KDIM + d;
        qr[o] = qv * cs + sgn * qp * sn;
        kr[o] = kv * cs + sgn * kp * sn;
        vvT[((size_t)bh * KDIM + d) * LL + l] = cv[r];  // transposed V for phase 3
    }
}

// -------- retention core: one workgroup per (b,h, 32-row q tile) --------
__global__ void __launch_bounds__(256) k_attn(
        const float* __restrict__ qr, const float* __restrict__ kr,
        const float* __restrict__ vvT, const float* __restrict__ rmask,
        float* __restrict__ attn) {
    __shared__ float S[32 * SROW];                      // 32 x 1024 scores, padded
    __shared__ float invden[32];
    __shared__ float partial[4 * 256];
    int w = threadIdx.x >> 5, lane = threadIdx.x & 31;
    int lh = lane >> 4, lr = lane & 15;
    int qt = blockIdx.x & 31, bh = blockIdx.x >> 5;
    int q0 = qt * 32;
    int h = bh & 7, b = bh >> 3;
    float dec = decay_of(h);
    const float* qb = qr  + (size_t)bh * LL * KDIM;
    const float* kb = kr  + (size_t)bh * LL * KDIM;
    const float* vb = vvT + (size_t)bh * KDIM * LL;

    // Phase 1: S = (Q Kt) * mask   (128 16x16 tiles over 8 waves)
    for (int t = w; t < 128; t += 8) {
        int qi = t >> 6, kt = t & 63;
        const float* arow = qb + (size_t)(q0 + qi * 16 + lr) * KDIM;
        const float* brow = kb + (size_t)(kt * 16 + lr) * KDIM;
        if (t + 8 < 128)                                // prefetch next K tile
            __builtin_prefetch(kb + (size_t)((((t + 8) & 63) * 16) + lr) * KDIM, 0, 1);
        v8f c = {};
        #pragma unroll
        for (int kk = 0; kk < KDIM; kk += 4) {
            v2f a   = *reinterpret_cast<const v2f*>(arow + kk + 2 * lh);
            v2f bfr = *reinterpret_cast<const v2f*>(brow + kk + 2 * lh);
            c = __builtin_amdgcn_wmma_f32_16x16x4_f32(false, a, false, bfr, (short)0, c, false, false);
        }
        int kg = kt * 16 + lr;
        int k1 = kg >> 5, k2 = kg & 31;
        #pragma unroll
        for (int r = 0; r < 8; ++r) {
            int ql = qi * 16 + r + 8 * lh;
            int qg = q0 + ql;
            float dd = (float)(iabs((qg >> 5) - k1) + iabs((qg & 31) - k2));
            float m = __expf(dec * dd) * rmask[h * LL + qg];
            S[ql * SROW + kg] = c[r] * m;
        }
    }
    __syncthreads();

    // Phase 2: 1/den per row (clip(sum |S|, 1, 5e4))
    #pragma unroll
    for (int rr = 0; rr < 4; ++rr) {
        int row = w + 8 * rr;
        float s = 0.0f;
        for (int j = 0; j < 32; ++j) s += fabsf(S[row * SROW + lane + 32 * j]);
        s = wave_reduce_add(s);
        if (lane == 0) invden[row] = 1.0f / fminf(fmaxf(s, 1.0f), 50000.0f);
    }
    __syncthreads();

    // Phase 3: O = S V, 4 output tiles, K split across wave pairs
    int t = w & 3, khalf = w >> 2;
    int ti = t >> 1, tj = t & 1;
    const float* vrow = vb + (size_t)(tj * 16 + lr) * LL;   // transposed V row
    v8f c = {};
    for (int k0 = khalf * 512; k0 < khalf * 512 + 512; k0 += 4) {
        int ko = k0 + 2 * lh;
        v2f a   = *reinterpret_cast<const v2f*>(&S[(ti * 16 + lr) * SROW + ko]);
        v2f bfr = *reinterpret_cast<const v2f*>(vrow + ko);
        c = __builtin_amdgcn_wmma_f32_16x16x4_f32(false, a, false, bfr, (short)0, c, false, false);
    }
    if (khalf == 1) {
        #pragma unroll
        for (int r = 0; r < 8; ++r)
            partial[t * 256 + (r + 8 * lh) * 16 + lr] = c[r];
    }
    __syncthreads();
    if (khalf == 0) {
        #pragma unroll
        for (int r = 0; r < 8; ++r) {
            int m = r + 8 * lh;
            float o = (c[r] + partial[t * 256 + m * 16 + lr]) * invden[ti * 16 + m];
            int l = q0 + ti * 16 + m;
            attn[((size_t)(b * LL + l)) * EE + h * KDIM + tj * 16 + lr] = o;
        }
    }
}

// -------- LayerNorm(32) + exact GELU, one wave per (b,l,h) --------
__global__ void k_ln_gelu(const float* __restrict__ attn,
                          const float* __restrict__ ln_g,
                          const float* __restrict__ ln_b,
                          float* __restrict__ act) {
    int w = threadIdx.x >> 5, lane = threadIdx.x & 31;
    size_t gid = (size_t)blockIdx.x * 8 + w;           // 65536 groups
    float v = attn[gid * 32 + lane];
    float mu = wave_reduce_add(v) * (1.0f / 32.0f);
    float dv = v - mu;
    float var = wave_reduce_add(dv * dv) * (1.0f / 32.0f);
    float nv = dv * rsqrtf(var + LN_EPS) * ln_g[lane] + ln_b[lane];
    act[gid * 32 + lane] = 0.5f * nv * (1.0f + erff(nv * 0.70710678118654752f));
}

// -------- final projection: out = act @ Wo^T (A tile staged via TDM) --------
__global__ void __launch_bounds__(256) k_out(const float* __restrict__ act,
                                             const float* __restrict__ Wo,
                                             float* __restrict__ out) {
    __shared__ float as[16 * EE];                       // 16KB A tile
    int w = threadIdx.x >> 5, lane = threadIdx.x & 31;
    int lh = lane >> 4, lr = lane & 15;
    int mt = blockIdx.x >> 1;
    int nt = (blockIdx.x & 1) * 8 + w;

    if (threadIdx.x < 32) {
        tdm_load_2d_f32((uint32_t)(uintptr_t)&as[0], act + (size_t)mt * 16 * EE,
                        EE, 16, EE);
        __builtin_amdgcn_s_wait_tensorcnt(0);
    }
    __syncthreads();

    const float* brow = Wo + (size_t)(nt * 16 + lr) * EE;
    v8f c = {};
    for (int k0 = 0; k0 < EE; k0 += 4) {
        int ko = k0 + 2 * lh;
        v2f a   = *reinterpret_cast<const v2f*>(&as[lr * EE + ko]);
        v2f bfr = *reinterpret_cast<const v2f*>(brow + ko);
        c = __builtin_amdgcn_wmma_f32_16x16x4_f32(false, a, false, bfr, (short)0, c, false, false);
    }
    #pragma unroll
    for (int r = 0; r < 8; ++r) {
        int row = mt * 16 + r + 8 * lh;
        out[(size_t)row * EE + nt * 16 + lr] = c[r];
    }
}

extern "C" void kernel_launch(void* const* d_in, const int* in_sizes, int n_in,
                              void* d_out, int out_size, void* d_ws, size_t ws_size,
                              hipStream_t stream) {
    const float* x    = (const float*)d_in[0];
    const float* Wq   = (const float*)d_in[1];
    const float* Wk   = (const float*)d_in[2];
    const float* Wv   = (const float*)d_in[3];
    const float* Wo   = (const float*)d_in[4];
    const float* ln_g = (const float*)d_in[5];
    const float* ln_b = (const float*)d_in[6];

    float* ws    = (float*)d_ws;
    float* qr    = ws;                                  // B*H*L*KD = 2M floats
    float* kr    = qr   + (size_t)BB * HH * LL * KDIM;
    float* vvT   = kr   + (size_t)BB * HH * LL * KDIM;  // transposed per (b,h)
    float* attn  = vvT  + (size_t)BB * HH * LL * KDIM;  // B*L*E
    float* act   = attn + (size_t)BB * LL * EE;
    float* sinT  = act  + (size_t)BB * LL * EE;
    float* cosT  = sinT + (size_t)LL * KDIM;
    float* rmask = cosT + (size_t)LL * KDIM;
    float* out   = (float*)d_out;

    k_tables <<<(LL * KDIM) / 256, 256, 0, stream>>>(sinT, cosT);
    k_rowsum <<<(HH * LL) / 8,     256, 0, stream>>>(rmask);
    k_proj   <<<(512 * 16) / 8,    256, 0, stream>>>(x, Wq, Wk, Wv, sinT, cosT, qr, kr, vvT);
    k_attn   <<<BB * HH * (LL/32), 256, 0, stream>>>(qr, kr, vvT, rmask, attn);
    k_ln_gelu<<<(BB * LL * HH)/8,  256, 0, stream>>>(attn, ln_g, ln_b, act);
    k_out    <<<(512 * 16) / 8,    256, 0, stream>>>(act, Wo, out);
}